// ConditionalVelocityModel_4612794876058
// MI455X (gfx1250) — compile-verified
//
#include <hip/hip_runtime.h>
#include <math.h>

typedef __attribute__((ext_vector_type(2))) float v2f;
typedef __attribute__((ext_vector_type(8))) float v8f;

#define B_   1024
#define S_   256
#define H_   128
#define COV_ 512
#define E_   8
#define ES_  7
#define PAD  130   // even (8B-aligned b64 LDS loads), 130 mod 64 = 2 (bank spread)
#define GB   16    // batch rows per routing block (one WMMA M-tile)
#define PADC 514   // cov row pitch in LDS
#define PADH 258   // hr row pitch in LDS

// Branchless, overflow-safe softplus: max(x,0) + log(1 + exp(-|x|)).
// v_exp_f32 / v_log_f32 hardware transcendentals; no divergent branches.
__device__ __forceinline__ float softplus_f(float x) {
  return fmaxf(x, 0.f) + __logf(1.f + __expf(-fabsf(x)));
}

// ---------------------------------------------------------------------------
// Routing / modulation precompute, 16 batch rows per block (grid = B/16).
// The two fat router GEMVs become real WMMA f32 GEMMs:
//   hr = relu(cov16[16,512] @ Wr1[512,256] + br1)
//   cg =       cov16[16,512] @ Wc [512,128] + bc
// Outputs per batch row b:
//   w_out[b][8], scale_out[b][128] = emb[t[b]]*(1+cg), beff_out[b][128] = w.b
// ---------------------------------------------------------------------------
struct RSMem {
  float cov_s[GB * PADC];   // ~32.9 KB
  float hr_s [GB * PADH];   // ~16.5 KB
  float cg_s [GB * PAD];    // ~8.3 KB
  float lg_s [GB * 8];
  float w_sh [GB * 8];
};

__global__ __launch_bounds__(256)
void routing16_kernel(const float* __restrict__ cov, const int* __restrict__ t,
                      const float* __restrict__ emb,   // [NSTEPS][H]
                      const float* __restrict__ bvec,  // [E][H]
                      const float* __restrict__ Wc, const float* __restrict__ bc,
                      const float* __restrict__ Wr1, const float* __restrict__ br1,
                      const float* __restrict__ Wr2, const float* __restrict__ br2,
                      float* __restrict__ w_out,
                      float* __restrict__ scale_out,
                      float* __restrict__ beff_out) {
  __shared__ RSMem sm;
  const int b0   = blockIdx.x * GB;
  const int tid  = threadIdx.x;
  const int wave = tid >> 5;
  const int lane = tid & 31;
  const int half = lane >> 4;
  const int li   = lane & 15;

  // stage cov16 [16][512] into LDS (coalesced)
  for (int idx = tid; idx < GB * COV_; idx += 256) {
    const int m = idx >> 9, c = idx & (COV_ - 1);
    sm.cov_s[m * PADC + c] = cov[(b0 + m) * COV_ + c];
  }
  __syncthreads();

  // ---- hr = relu(cov16 @ Wr1 + br1) : each wave owns 2 N-tiles of 16 ----
  {
    v8f acc[2];
    #pragma unroll
    for (int nt2 = 0; nt2 < 2; ++nt2) {
      const float bv = br1[(2 * wave + nt2) * 16 + li];
      #pragma unroll
      for (int r = 0; r < 8; ++r) acc[nt2][r] = bv;
    }
    for (int k = 0; k < COV_; k += 4) {
      const int ko = k + 2 * half;
      v2f a = *(const v2f*)&sm.cov_s[li * PADC + ko];
      #pragma unroll
      for (int nt2 = 0; nt2 < 2; ++nt2) {
        const int n = (2 * wave + nt2) * 16 + li;
        v2f bf;
        bf[0] = Wr1[ ko      * 256 + n];
        bf[1] = Wr1[(ko + 1) * 256 + n];
        acc[nt2] = __builtin_amdgcn_wmma_f32_16x16x4_f32(
            false, a, false, bf, (short)0, acc[nt2], false, false);
      }
    }
    #pragma unroll
    for (int nt2 = 0; nt2 < 2; ++nt2) {
      const int n = (2 * wave + nt2) * 16 + li;
      #pragma unroll
      for (int r = 0; r < 8; ++r) {
        const int m = r + 8 * half;          // C/D: VGPR r -> M = r (+8 hi half)
        sm.hr_s[m * PADH + n] = fmaxf(acc[nt2][r], 0.f);
      }
    }
  }
  __syncthreads();

  // ---- router logits [16][7] + softmax ----
  if (tid < GB * ES_) {
    const int m = tid / ES_, e = tid % ES_;
    float lg = br2[e];
    for (int k = 0; k < 256; ++k) lg = fmaf(sm.hr_s[m * PADH + k], Wr2[k * ES_ + e], lg);
    sm.lg_s[m * 8 + e] = lg;
  }
  __syncthreads();
  if (tid < GB * ES_) {
    const int m = tid / ES_, e = tid % ES_;
    float mx = sm.lg_s[m * 8];
    #pragma unroll
    for (int ee = 1; ee < ES_; ++ee) mx = fmaxf(mx, sm.lg_s[m * 8 + ee]);
    float sum = 0.f;
    #pragma unroll
    for (int ee = 0; ee < ES_; ++ee) sum += expf(sm.lg_s[m * 8 + ee] - mx);
    const float rw = expf(sm.lg_s[m * 8 + e] - mx) / sum * 0.7f;  // * (1-UW)
    w_out[(b0 + m) * E_ + 1 + e] = rw;
    sm.w_sh[m * 8 + 1 + e] = rw;
  }
  if (tid < GB) {                 // universal expert: UW/EU = 0.3
    w_out[(b0 + tid) * E_] = 0.3f;
    sm.w_sh[tid * 8] = 0.3f;
  }
  __syncthreads();

  // ---- cg = cov16 @ Wc + bc : each wave owns 1 N-tile of 16 ----
  {
    v8f acc;
    const int n = wave * 16 + li;
    const float bv = bc[n];
    #pragma unroll
    for (int r = 0; r < 8; ++r) acc[r] = bv;
    for (int k = 0; k < COV_; k += 4) {
      const int ko = k + 2 * half;
      v2f a = *(const v2f*)&sm.cov_s[li * PADC + ko];
      v2f bf;
      bf[0] = Wc[ ko      * H_ + n];
      bf[1] = Wc[(ko + 1) * H_ + n];
      acc = __builtin_amdgcn_wmma_f32_16x16x4_f32(
          false, a, false, bf, (short)0, acc, false, false);
    }
    #pragma unroll
    for (int r = 0; r < 8; ++r) {
      const int m = r + 8 * half;
      sm.cg_s[m * PAD + n] = acc[r];
    }
  }
  __syncthreads();

  // ---- scale & beff for the 16 rows ----
  for (int idx = tid; idx < GB * H_; idx += 256) {
    const int m = idx >> 7, j = idx & (H_ - 1);
    const int b = b0 + m;
    const float gamma = emb[t[b] * H_ + j];
    scale_out[b * H_ + j] = gamma * (1.f + sm.cg_s[m * PAD + j]);
    float be = 0.f;
    #pragma unroll
    for (int e = 0; e < E_; ++e) be = fmaf(sm.w_sh[m * 8 + e], bvec[e * H_ + j], be);
    beff_out[b * H_ + j] = be;
  }
}

// ---------------------------------------------------------------------------
// Fused 3-layer MoE MLP for one batch row. grid = B, block = 256 (8 wave32).
// Activations [256][128] fp32 live entirely in LDS (CDNA5 320KB WGP LDS).
// Layers 2/3: WMMA F32 16x16x4 GEMMs vs per-batch effective weight in LDS.
// ---------------------------------------------------------------------------
struct SMem {
  float hA[S_ * PAD];   // activations, row-major [s][h]
  float WT[H_ * PAD];   // effective weight, transposed [n][k]
  float xs0[S_];
  float xs1[S_];
  float w_s[E_];
  float beff_s[H_];
  float scale_s[H_];
  float we0[H_];
  float we1[H_];
  float w4_s[H_];
};

__global__ __launch_bounds__(256)
void moe_fused_kernel(const float* __restrict__ y_t, const float* __restrict__ y0,
                      const float* __restrict__ W1,   // [8][2][128]
                      const float* __restrict__ W2,   // [8][128][128]
                      const float* __restrict__ W3,   // [8][128][128]
                      const float* __restrict__ W4,   // [128][1]
                      const float* __restrict__ b4,   // [1]
                      const float* __restrict__ w_all,     // [3][B][8]
                      const float* __restrict__ scale_all, // [3][B][128]
                      const float* __restrict__ beff_all,  // [3][B][128]
                      float* __restrict__ out) {
  __shared__ SMem sm;
  const int b    = blockIdx.x;
  const int tid  = threadIdx.x;
  const int wave = tid >> 5;
  const int lane = tid & 31;
  const int half = lane >> 4;   // K-half of the wave (ISA A/B f32 layout)
  const int li   = lane & 15;

  sm.xs0[tid] = y_t[b * S_ + tid];
  sm.xs1[tid] = y0 [b * S_ + tid];
  if (tid < H_) sm.w4_s[tid] = W4[tid];

  // ---------------- layer 1 (K = 2, rank-2 VALU update) ----------------
  if (tid < E_) sm.w_s[tid] = w_all[(0 * B_ + b) * E_ + tid];
  if (tid < H_) {
    sm.beff_s[tid]  = beff_all [(0 * B_ + b) * H_ + tid];
    sm.scale_s[tid] = scale_all[(0 * B_ + b) * H_ + tid];
  }
  __syncthreads();
  if (tid < H_) {
    float a0 = 0.f, a1 = 0.f;
    #pragma unroll
    for (int e = 0; e < E_; ++e) {
      a0 = fmaf(sm.w_s[e], W1[e * 2 * H_ + tid],      a0);
      a1 = fmaf(sm.w_s[e], W1[e * 2 * H_ + H_ + tid], a1);
    }
    sm.we0[tid] = a0;
    sm.we1[tid] = a1;
  }
  __syncthreads();
  {
    const int j  = tid & (H_ - 1);
    const int s0 = tid >> 7;
    const float w0 = sm.we0[j], w1 = sm.we1[j];
    const float be = sm.beff_s[j], sc = sm.scale_s[j];
    for (int s = s0; s < S_; s += 2) {
      float v = (fmaf(sm.xs0[s], w0, sm.xs1[s] * w1) + be) * sc;
      sm.hA[s * PAD + j] = softplus_f(v);
    }
  }
  __syncthreads();

  // ---------------- layers 2 and 3 (WMMA f32 GEMM) ----------------
  for (int l = 1; l <= 2; ++l) {
    const float* __restrict__ Wl = (l == 1) ? W2 : W3;
    if (tid < E_) sm.w_s[tid] = w_all[(l * B_ + b) * E_ + tid];
    if (tid < H_) {
      sm.beff_s[tid]  = beff_all [(l * B_ + b) * H_ + tid];
      sm.scale_s[tid] = scale_all[(l * B_ + b) * H_ + tid];
    }
    __syncthreads();

    // WeffT[n][k] = sum_e w[e] * Wl[e][k][n]  (coalesced along n)
    for (int idx = tid; idx < H_ * H_; idx += 256) {
      const int n = idx & (H_ - 1);
      const int k = idx >> 7;
      float acc = 0.f;
      #pragma unroll
      for (int e = 0; e < E_; ++e)
        acc = fmaf(sm.w_s[e], Wl[(e * H_ + k) * H_ + n], acc);
      sm.WT[n * PAD + k] = acc;
    }
    __syncthreads();

    // Each wave: 2 M-tiles x 8 N-tiles of 16x16, K-loop of 32 x (16x16x4).
    v8f acc[2][8];
    #pragma unroll
    for (int mt = 0; mt < 2; ++mt)
      #pragma unroll
      for (int nt = 0; nt < 8; ++nt) {
        const float bv = sm.beff_s[nt * 16 + li];
        #pragma unroll
        for (int r = 0; r < 8; ++r) acc[mt][nt][r] = bv;
      }

    const int m0 = 32 * wave + li;   // A rows: M = li in both wave halves
    for (int k = 0; k < H_; k += 4) {
      const int ko = k + 2 * half;   // low half lanes: K,K+1 ; high: K+2,K+3
      v2f a0 = *(const v2f*)&sm.hA[ m0       * PAD + ko];
      v2f a1 = *(const v2f*)&sm.hA[(m0 + 16) * PAD + ko];
      v2f bf[8];
      #pragma unroll
      for (int nt = 0; nt < 8; ++nt)
        bf[nt] = *(const v2f*)&sm.WT[(nt * 16 + li) * PAD + ko];
      #pragma unroll
      for (int nt = 0; nt < 8; ++nt) {
        acc[0][nt] = __builtin_amdgcn_wmma_f32_16x16x4_f32(
            false, a0, false, bf[nt], (short)0, acc[0][nt], false, false);
        acc[1][nt] = __builtin_amdgcn_wmma_f32_16x16x4_f32(
            false, a1, false, bf[nt], (short)0, acc[1][nt], false, false);
      }
    }
    __syncthreads();   // all waves done reading hA before overwrite

    // scale + softplus fused into the LDS writeback (branchless).
    #pragma unroll
    for (int mt = 0; mt < 2; ++mt) {
      const int mb = 32 * wave + 16 * mt + 8 * half;  // C/D: VGPR r -> M=r(+8)
      #pragma unroll
      for (int nt = 0; nt < 8; ++nt) {
        const int n  = nt * 16 + li;
        const float sc = sm.scale_s[n];
        #pragma unroll
        for (int r = 0; r < 8; ++r) {
          float v = acc[mt][nt][r] * sc;
          sm.hA[(mb + r) * PAD + n] = softplus_f(v);
        }
      }
    }
    __syncthreads();
  }

  // ---------------- output head: h @ W4 + b4 ----------------
  {
    float s = 0.f;
    #pragma unroll 8
    for (int hh = 0; hh < H_; ++hh)
      s = fmaf(sm.hA[tid * PAD + hh], sm.w4_s[hh], s);
    out[b * S_ + tid] = s + b4[0];
  }
}

// ---------------------------------------------------------------------------
// Load-balance loss: sum_l mean_e (mean_b rw - 1/7)^2 -> d_out[B*S]
// ---------------------------------------------------------------------------
__global__ void finalize_kernel(const float* __restrict__ w_all, float* __restrict__ out_load) {
  __shared__ float us[3 * ES_];
  const int tid = threadIdx.x;
  if (tid < 3 * ES_) {
    const int l = tid / ES_, e = tid % ES_;
    float s = 0.f;
    for (int b = 0; b < B_; ++b) s += w_all[(l * B_ + b) * E_ + 1 + e];
    us[tid] = s * (1.f / B_);
  }
  __syncthreads();
  if (tid == 0) {
    float total = 0.f;
    for (int l = 0; l < 3; ++l) {
      float ld = 0.f;
      for (int e = 0; e < ES_; ++e) {
        float d = us[l * ES_ + e] - (1.f / 7.f);
        ld += d * d;
      }
      total += ld * (1.f / 7.f);
    }
    out_load[0] = total;
  }
}

// ---------------------------------------------------------------------------
extern "C" void kernel_launch(void* const* d_in, const int* in_sizes, int n_in,
                              void* d_out, int out_size, void* d_ws, size_t ws_size,
                              hipStream_t stream) {
  const float* y_t = (const float*)d_in[0];
  const float* y0  = (const float*)d_in[1];
  const float* cov = (const float*)d_in[2];
  const int*   t   = (const int*)  d_in[3];
  const float* W4  = (const float*)d_in[31];
  const float* b4  = (const float*)d_in[32];

  // workspace layout (floats)
  float* ws        = (float*)d_ws;
  float* w_all     = ws;                          // [3][B][8]
  float* scale_all = ws + 3 * B_ * E_;            // [3][B][128]
  float* beff_all  = scale_all + 3 * B_ * H_;     // [3][B][128]

  const float* Wexp[3];
  for (int l = 0; l < 3; ++l) {
    const int base = 4 + 9 * l;
    const float* emb = (const float*)d_in[base + 0];
    Wexp[l]          = (const float*)d_in[base + 1];
    const float* bv  = (const float*)d_in[base + 2];
    const float* Wc  = (const float*)d_in[base + 3];
    const float* bc  = (const float*)d_in[base + 4];
    const float* Wr1 = (const float*)d_in[base + 5];
    const float* br1 = (const float*)d_in[base + 6];
    const float* Wr2 = (const float*)d_in[base + 7];
    const float* br2 = (const float*)d_in[base + 8];
    routing16_kernel<<<B_ / GB, 256, 0, stream>>>(
        cov, t, emb, bv, Wc, bc, Wr1, br1, Wr2, br2,
        w_all + l * B_ * E_, scale_all + l * B_ * H_, beff_all + l * B_ * H_);
  }

  float* out = (float*)d_out;
  moe_fused_kernel<<<B_, 256, 0, stream>>>(
      y_t, y0, Wexp[0], Wexp[1], Wexp[2], W4, b4,
      w_all, scale_all, beff_all, out);

  finalize_kernel<<<1, 32, 0, stream>>>(w_all, out + B_ * S_);
}